// Cheb1_84954453114993
// MI455X (gfx1250) — compile-verified
//
#include <hip/hip_runtime.h>

typedef __attribute__((ext_vector_type(2))) float v2f;
typedef __attribute__((ext_vector_type(4))) float v4f;
typedef __attribute__((ext_vector_type(8))) float v8f;

#define N_NODES 100000
#define N_EDGES 1600000
#define D 64

// ---------------------------------------------------------------------------
// Kernel 1: zero deg[] and Tx1[] in workspace (harness poisons ws with 0xAA).
// ---------------------------------------------------------------------------
__global__ __launch_bounds__(256) void zero_ws_kernel(float* __restrict__ deg,
                                                      float* __restrict__ tx1) {
    int i = blockIdx.x * 256 + threadIdx.x;
    if (i < N_NODES) deg[i] = 0.0f;
    if (i < N_NODES * D) tx1[i] = 0.0f;
}

// ---------------------------------------------------------------------------
// Kernel 2: deg[row[e]] += w[e]   (native global_atomic_add_f32, L2-resident)
// ---------------------------------------------------------------------------
__global__ __launch_bounds__(256) void deg_kernel(const int* __restrict__ ei,
                                                  const float* __restrict__ ew,
                                                  float* __restrict__ deg) {
    int e = blockIdx.x * 256 + threadIdx.x;
    if (e >= N_EDGES) return;
    unsafeAtomicAdd(&deg[ei[e]], ew[e]);
}

// ---------------------------------------------------------------------------
// Kernel 3: dinv[i] = deg[i] > 0 ? rsqrt(deg[i]) : 0
// ---------------------------------------------------------------------------
__global__ __launch_bounds__(256) void dinv_kernel(const float* __restrict__ deg,
                                                   float* __restrict__ dinv) {
    int i = blockIdx.x * 256 + threadIdx.x;
    if (i >= N_NODES) return;
    float d = deg[i];
    dinv[i] = (d > 0.0f) ? __frsqrt_rn(d) : 0.0f;
}

// ---------------------------------------------------------------------------
// Kernel 4: edge scatter. 16 threads per edge, each thread owns 4 features:
//   Tx1[col,:] += (-dinv[row]*w*dinv[col]) * x[row,:]
// float4 gather of x (16B, 16-aligned) + 4 native f32 atomics. x and Tx1 both
// fit in the 192MB L2, so this is an L2-rate atomic workload, not HBM-bound.
// ---------------------------------------------------------------------------
__global__ __launch_bounds__(256) void scatter_kernel(const float* __restrict__ x,
                                                      const int* __restrict__ ei,
                                                      const float* __restrict__ ew,
                                                      const float* __restrict__ dinv,
                                                      float* __restrict__ tx1) {
    int g = blockIdx.x * 256 + threadIdx.x;   // E*16 = 25.6M threads total
    int e = g >> 4;
    int f = (g & 15) * 4;
    if (e >= N_EDGES) return;
    int row = ei[e];
    int col = ei[N_EDGES + e];
    float norm = -dinv[row] * ew[e] * dinv[col];
    v4f xv = *reinterpret_cast<const v4f*>(x + (long long)row * D + f);
    float* dst = tx1 + (long long)col * D + f;
    unsafeAtomicAdd(dst + 0, norm * xv.x);
    unsafeAtomicAdd(dst + 1, norm * xv.y);
    unsafeAtomicAdd(dst + 2, norm * xv.z);
    unsafeAtomicAdd(dst + 3, norm * xv.w);
}

// ---------------------------------------------------------------------------
// Kernel 5: out = x @ W0 + Tx1 @ W1 + b via V_WMMA_F32_16X16X4_F32.
// 8 waves/block, one wave per 16-row node tile (100000/16 = 6250 tiles).
// W0, W1, b staged in LDS. Per wave: 4 column-tile accumulators (v8f each),
// K=64 swept in 16 steps of K=4 -> 128 wmma per wave. Bias folded into C init.
//
// ISA layouts (CDNA5 ISA):
//   A 16x4 f32 : lane(0-15)  v0=A[m][k0+0] v1=A[m][k0+1], m=lane
//                lane(16-31) v0=A[m][k0+2] v1=A[m][k0+3], m=lane-16
//   B 4x16 f32 : mirrored (k rows striped across the 2 VGPRs / lane halves)
//   C/D 16x16  : row = 8*(lane>>4) + vgpr, col = lane&15
// ---------------------------------------------------------------------------
__global__ __launch_bounds__(256) void cheb_gemm_kernel(const float* __restrict__ x,
                                                        const float* __restrict__ tx1,
                                                        const float* __restrict__ W0,
                                                        const float* __restrict__ W1,
                                                        const float* __restrict__ b,
                                                        float* __restrict__ out) {
    __shared__ float sW0[D * D];
    __shared__ float sW1[D * D];
    __shared__ float sb[D];

    int tid = threadIdx.x;
    for (int i = tid; i < D * D; i += 256) {
        sW0[i] = W0[i];
        sW1[i] = W1[i];
    }
    if (tid < D) sb[tid] = b[tid];
    __syncthreads();

    int wave = tid >> 5;
    int lane = tid & 31;
    int m0 = (blockIdx.x * 8 + wave) * 16;
    if (m0 >= N_NODES) return;

    int half = lane >> 4;   // 0: K={0,1} slice, 1: K={2,3} slice
    int mrow = lane & 15;

    v8f acc[4];
#pragma unroll
    for (int n = 0; n < 4; ++n) {
        float bias = sb[n * 16 + mrow];   // bias depends only on output col
#pragma unroll
        for (int r = 0; r < 8; ++r) acc[n][r] = bias;
    }

    const float* xrow = x   + (long long)(m0 + mrow) * D;
    const float* trow = tx1 + (long long)(m0 + mrow) * D;

#pragma unroll
    for (int ks = 0; ks < 16; ++ks) {
        int kk = ks * 4 + 2 * half;
        v2f ax = *reinterpret_cast<const v2f*>(xrow + kk);   // A frag (x tile)
        v2f at = *reinterpret_cast<const v2f*>(trow + kk);   // A frag (Tx1 tile)
#pragma unroll
        for (int n = 0; n < 4; ++n) {
            int col = n * 16 + mrow;
            v2f b0, b1;
            b0.x = sW0[kk * D + col];
            b0.y = sW0[(kk + 1) * D + col];
            b1.x = sW1[kk * D + col];
            b1.y = sW1[(kk + 1) * D + col];
            acc[n] = __builtin_amdgcn_wmma_f32_16x16x4_f32(
                false, ax, false, b0, (short)0, acc[n], false, false);
            acc[n] = __builtin_amdgcn_wmma_f32_16x16x4_f32(
                false, at, false, b1, (short)0, acc[n], false, false);
        }
    }

#pragma unroll
    for (int n = 0; n < 4; ++n) {
        int col = n * 16 + mrow;
#pragma unroll
        for (int r = 0; r < 8; ++r) {
            int row = m0 + half * 8 + r;
            out[(long long)row * D + col] = acc[n][r];
        }
    }
}

// ---------------------------------------------------------------------------
// Launcher
// ---------------------------------------------------------------------------
extern "C" void kernel_launch(void* const* d_in, const int* in_sizes, int n_in,
                              void* d_out, int out_size, void* d_ws, size_t ws_size,
                              hipStream_t stream) {
    const float* x  = (const float*)d_in[0];
    const int*   ei = (const int*)d_in[1];     // edge_index [2, E] flat
    const float* ew = (const float*)d_in[2];
    const float* W0 = (const float*)d_in[3];
    const float* W1 = (const float*)d_in[4];
    const float* b  = (const float*)d_in[5];
    float* out = (float*)d_out;

    // workspace layout (floats): deg[N] | dinv[N] | Tx1[N*D]  = 26.4 MB
    float* deg  = (float*)d_ws;
    float* dinv = deg + N_NODES;
    float* tx1  = deg + 2 * (size_t)N_NODES;

    zero_ws_kernel<<<(N_NODES * D + 255) / 256, 256, 0, stream>>>(deg, tx1);
    deg_kernel<<<(N_EDGES + 255) / 256, 256, 0, stream>>>(ei, ew, deg);
    dinv_kernel<<<(N_NODES + 255) / 256, 256, 0, stream>>>(deg, dinv);
    scatter_kernel<<<(N_EDGES * 16) / 256, 256, 0, stream>>>(x, ei, ew, dinv, tx1);
    cheb_gemm_kernel<<<((N_NODES / 16) + 7) / 8, 256, 0, stream>>>(x, tx1, W0, W1, b, out);
}